// SmoothAP_50379966382605
// MI455X (gfx1250) — compile-verified
//
#include <hip/hip_runtime.h>
#include <hip/hip_bf16.h>

typedef __attribute__((ext_vector_type(2))) float v2f;
typedef __attribute__((ext_vector_type(8))) float v8f;

#define BATCH 512
#define FEAT  128
#define GRP   8          // g = BATCH / NUM_ID
#define INV_GB (1.0f / 4096.0f)   // 1/(g*B)

// temp-scaled sigmoid: 1/(1+exp(clip(-x/0.01, -50, 50)))
__device__ __forceinline__ float tsig(float x) {
    float e = fminf(fmaxf(-100.0f * x, -50.0f), 50.0f);
    return 1.0f / (1.0f + __expf(e));
}

// ---------------------------------------------------------------------------
// Kernel A: S = P * P^T via V_WMMA_F32_16X16X4_F32.
// One wave per 16x16 output tile; 4 waves per block; 32x32 tiles total.
// A fragment (16x4 f32): lane L holds M = L%16, VGPRs {K = 2*(L/16), 2*(L/16)+1}
// B fragment (4x16 f32): lane L holds N = L%16, same K pattern -> since
// B = P^T, both fragments are contiguous float2 loads from rows of P.
// ---------------------------------------------------------------------------
__global__ void __launch_bounds__(128)
smoothap_gemm_wmma(const float* __restrict__ P, float* __restrict__ S) {
    const int wave = threadIdx.x >> 5;
    const int lane = threadIdx.x & 31;
    const int tile = blockIdx.x * 4 + wave;       // 0..1023
    const int tileRow = tile >> 5;                // 0..31
    const int tileCol = tile & 31;                // 0..31
    const int row0 = tileRow << 4;
    const int col0 = tileCol << 4;

    const int m    = lane & 15;
    const int half = lane >> 4;

    const float* __restrict__ arow = P + (size_t)(row0 + m) * FEAT + 2 * half;
    const float* __restrict__ brow = P + (size_t)(col0 + m) * FEAT + 2 * half;

    v8f acc = {};
#pragma unroll
    for (int kk = 0; kk < FEAT; kk += 4) {
        v2f a = *(const v2f*)(arow + kk);   // K = kk+2*half, kk+2*half+1
        v2f b = *(const v2f*)(brow + kk);
        acc = __builtin_amdgcn_wmma_f32_16x16x4_f32(
                  /*neg_a=*/false, a, /*neg_b=*/false, b,
                  /*c_mod=*/(short)0, acc,
                  /*reuse_a=*/false, /*reuse_b=*/false);
    }

    // C layout: VGPR i -> M = i + 8*(lane/16), N = lane%16
    const int n_out = lane & 15;
#pragma unroll
    for (int i = 0; i < 8; ++i) {
        int m_out = i + 8 * half;
        S[(size_t)(row0 + m_out) * BATCH + col0 + n_out] = acc[i];
    }
}

// ---------------------------------------------------------------------------
// Kernel B: per row r, compute partial[r] = sum_j pos_rk[r,j] / all_rk[r,j]
// where j runs over the 8 in-group columns.
// Block = 256 threads = 8 waves; wave w handles column j=w.
// all_rk = sum_k sigmoid(S[r,k]-S[r,j]) + 0.5   (k==j term is exactly 0.5,
// masked out, then +1: net +0.5). Same trick for pos_rk over the 8 in-group k.
// ---------------------------------------------------------------------------
__global__ void __launch_bounds__(256)
smoothap_rank(const float* __restrict__ S, float* __restrict__ partial) {
    const int r    = blockIdx.x;          // 0..511
    const int grp  = r >> 3;
    const int base = grp << 3;            // first column of r's group
    const int jb   = threadIdx.x >> 5;    // wave id -> in-group column 0..7
    const int lane = threadIdx.x & 31;

    const float* __restrict__ Srow = S + (size_t)r * BATCH;
    const float sdiag = Srow[base + jb];

    float sum = 0.0f;
#pragma unroll 4
    for (int k = lane; k < BATCH; k += 32) {
        sum += tsig(Srow[k] - sdiag);
    }
    // wave32 butterfly reduction
#pragma unroll
    for (int off = 16; off > 0; off >>= 1) {
        sum += __shfl_xor(sum, off);
    }

    __shared__ float allrk[GRP];
    if (lane == 0) allrk[jb] = sum + 0.5f;   // (-0.5 mask) + 1.0
    __syncthreads();

    if (threadIdx.x == 0) {
        float sg[GRP];
#pragma unroll
        for (int k = 0; k < GRP; ++k) sg[k] = Srow[base + k];
        float acc = 0.0f;
#pragma unroll
        for (int j = 0; j < GRP; ++j) {
            float pos = 0.0f;
#pragma unroll
            for (int k = 0; k < GRP; ++k) pos += tsig(sg[k] - sg[j]);
            pos += 0.5f;                      // (-0.5 mask) + 1.0
            acc += pos / allrk[j];
        }
        partial[r] = acc;
    }
}

// ---------------------------------------------------------------------------
// Kernel C: deterministic single-block tree reduction of 512 partials,
// out[0] = 1 - sum/4096.
// ---------------------------------------------------------------------------
__global__ void __launch_bounds__(256)
smoothap_finish(const float* __restrict__ partial, float* __restrict__ out) {
    __shared__ float red[256];
    const int t = threadIdx.x;
    red[t] = partial[t] + partial[t + 256];
    __syncthreads();
#pragma unroll
    for (int stride = 128; stride > 0; stride >>= 1) {
        if (t < stride) red[t] += red[t + stride];
        __syncthreads();
    }
    if (t == 0) out[0] = 1.0f - red[0] * INV_GB;
}

extern "C" void kernel_launch(void* const* d_in, const int* in_sizes, int n_in,
                              void* d_out, int out_size, void* d_ws, size_t ws_size,
                              hipStream_t stream) {
    (void)in_sizes; (void)n_in; (void)out_size; (void)ws_size;
    const float* preds = (const float*)d_in[0];
    float* out = (float*)d_out;

    float* S       = (float*)d_ws;                 // 512*512 floats = 1 MB
    float* partial = S + (size_t)BATCH * BATCH;    // 512 floats

    // 1024 tiles, 4 waves (tiles) per 128-thread block -> 256 blocks
    smoothap_gemm_wmma<<<dim3(256), dim3(128), 0, stream>>>(preds, S);
    // one block per row
    smoothap_rank<<<dim3(BATCH), dim3(256), 0, stream>>>(S, partial);
    // final deterministic reduction
    smoothap_finish<<<dim3(1), dim3(256), 0, stream>>>(partial, out);
}